// FractalDimension_31885837205966
// MI455X (gfx1250) — compile-verified
//
#include <hip/hip_runtime.h>
#include <stdint.h>

// out[r] = sum_l in[r, l] * (1/(l+1)),  r in [0, C*B*N), L = 1024.
// Memory-bound GEMV mapped onto V_WMMA_F32_16X16X4_F32 with
// global_load_async_to_lds_b128 streaming (double-buffered, wave-private).

typedef float v2f __attribute__((ext_vector_type(2)));
typedef float v8f __attribute__((ext_vector_type(8)));

#define L_DIM 1024
#define KC 32                               // k-chunk staged per buffer
#define NCHUNK (L_DIM / KC)                 // 32
#define PITCH 36                            // LDS row pitch (floats): 144B rows, 16B aligned, conflict-free
#define ROWS_PER_WAVE 16
#define WAVES_PER_BLOCK 8
#define ROWS_PER_BLOCK (ROWS_PER_WAVE * WAVES_PER_BLOCK)   // 128
#define WAVE_BUF_FLOATS (ROWS_PER_WAVE * PITCH)            // 576 floats = 2304 B per buffer
#define A_FLOATS (WAVES_PER_BLOCK * 2 * WAVE_BUF_FLOATS)   // 9216 floats = 36 KiB

__device__ __forceinline__ void async_load_b128(uint32_t lds_off_bytes,
                                                uint32_t g_off_bytes,
                                                const float* sbase) {
  // GVS mode: global address is the 64b SGPR pair in saddr plus the 32b offset in vaddr; LDS address is in vdst
  asm volatile("global_load_async_to_lds_b128 %0, %1, %2"
               :: "v"(lds_off_bytes), "v"(g_off_bytes), "s"(sbase)
               : "memory");
}
__device__ __forceinline__ void wait_async4() {
  asm volatile("s_wait_asynccnt 0x4" ::: "memory");
}
__device__ __forceinline__ void wait_async0() {
  asm volatile("s_wait_asynccnt 0x0" ::: "memory");
}

__global__ void __launch_bounds__(256)
fractal_rowdot_wmma(const float* __restrict__ in, float* __restrict__ out, int rows) {
  __shared__ __align__(16) float sA[A_FLOATS];   // per-wave double-buffered A tiles
  __shared__ __align__(16) float sW[L_DIM];      // weights 1/(l+1)

  const int tid = threadIdx.x;
  for (int i = tid; i < L_DIM; i += 256) sW[i] = 1.0f / (float)(i + 1);
  __syncthreads();

  const int wave = tid >> 5;
  const int lane = tid & 31;
  const int rowBase = blockIdx.x * ROWS_PER_BLOCK + wave * ROWS_PER_WAVE;
  if (rowBase + ROWS_PER_WAVE > rows) return;    // uniform per wave; no barriers follow

  const float* gbase = in + (size_t)rowBase * L_DIM;

  float* waveA = sA + wave * 2 * WAVE_BUF_FLOATS;
  const uint32_t ldsBuf0 = (uint32_t)(uintptr_t)waveA;       // LDS byte offset (low 32b of flat addr)
  const uint32_t ldsBuf1 = ldsBuf0 + WAVE_BUF_FLOATS * 4;

  // Per-lane async-copy coordinates: chunk = 16 rows x 32 floats = 128 chunks of 16B,
  // covered by 4 b128 instructions x 32 lanes.
  uint32_t g0[4], l0[4];
#pragma unroll
  for (int i = 0; i < 4; ++i) {
    const int idx = i * 32 + lane;
    const int r = idx >> 3;           // row within wave tile (0..15)
    const int ch = idx & 7;           // 16B chunk within the 128B row slice
    g0[i] = (uint32_t)(r * (L_DIM * 4) + ch * 16);
    l0[i] = (uint32_t)(r * (PITCH * 4) + ch * 16);
  }

  // Prologue: stage chunk 0 into buffer 0.
#pragma unroll
  for (int i = 0; i < 4; ++i) async_load_b128(ldsBuf0 + l0[i], g0[i], gbase);

  // WMMA fragment addressing.
  // A (16x4 f32): lane L: M = L%16; VGPR0/1 hold K = {0,1} (lanes 0-15) or {2,3} (lanes 16-31).
  // B (4x16 f32): broadcast w across N; lanes 0-15 need w[k..k+1], lanes 16-31 w[k+2..k+3].
  const int m = lane & 15;
  const int offK = (lane & 16) ? 2 : 0;
  const float* aBase0 = waveA + m * PITCH + offK;
  const float* aBase1 = aBase0 + WAVE_BUF_FLOATS;

  v8f acc = {};   // D/C: VGPR j = row (j | (lane>=16 ? 8 : 0)), all N columns identical

  for (int c = 0; c < NCHUNK; ++c) {
    if (c + 1 < NCHUNK) {
      const uint32_t kNextBytes = (uint32_t)(c + 1) * KC * 4;
      const uint32_t buf = ((c + 1) & 1) ? ldsBuf1 : ldsBuf0;
#pragma unroll
      for (int i = 0; i < 4; ++i) async_load_b128(buf + l0[i], g0[i] + kNextBytes, gbase);
      wait_async4();                 // in-order completion => chunk c is resident
    } else {
      wait_async0();
    }
    const float* aB = (c & 1) ? aBase1 : aBase0;
    const float* wB = sW + c * KC + offK;
#pragma unroll
    for (int k0 = 0; k0 < KC; k0 += 4) {
      v2f a = *(const v2f*)(aB + k0);      // ds_load_b64, conflict-free (pitch 36)
      v2f b = *(const v2f*)(wB + k0);      // ds_load_b64, same-address broadcast
      acc = __builtin_amdgcn_wmma_f32_16x16x4_f32(
          /*neg_a=*/false, a, /*neg_b=*/false, b,
          /*c_mod=*/(short)0, acc, /*reuse_a=*/false, /*reuse_b=*/false);
    }
  }

  // Lane 0 holds rows 0..7 of the tile in acc[0..7]; lane 16 holds rows 8..15.
  if (lane == 0 || lane == 16) {
    float* o = out + rowBase + ((lane >> 4) * 8);
#pragma unroll
    for (int j = 0; j < 8; ++j) o[j] = acc[j];
  }
}

extern "C" void kernel_launch(void* const* d_in, const int* in_sizes, int n_in,
                              void* d_out, int out_size, void* d_ws, size_t ws_size,
                              hipStream_t stream) {
  (void)n_in; (void)d_ws; (void)ws_size; (void)out_size;
  const float* in = (const float*)d_in[0];
  float* out = (float*)d_out;
  const int rows = in_sizes[0] / L_DIM;                    // 131072 for [4,8,4096,1024]
  const int blocks = (rows + ROWS_PER_BLOCK - 1) / ROWS_PER_BLOCK;  // 1024
  hipLaunchKernelGGL(fractal_rowdot_wmma, dim3(blocks), dim3(256), 0, stream,
                     in, out, rows);
}